// PartialRadonStep_54408645705958
// MI455X (gfx1250) — compile-verified
//
#include <hip/hip_runtime.h>
#include <math.h>

#define RW 256           // image width/height
#define NANG 180
#define ANG_PER_BLOCK 4  // 45 * 4 = 180
#define BLOCK 256        // 8 waves of 32

typedef float v2f __attribute__((ext_vector_type(2)));
typedef float v8f __attribute__((ext_vector_type(8)));
typedef int   v4i __attribute__((ext_vector_type(4)));

typedef __attribute__((address_space(1))) v4i g4_t;   // global 16B vector
typedef __attribute__((address_space(3))) v4i l4_t;   // LDS    16B vector
typedef __attribute__((address_space(3))) char lch_t;

// ---- one zero-padded tap from the LDS image -------------------------------
__device__ __forceinline__ float lds_tap(const float* sm, float xf, float yf) {
    bool vx = (xf >= 0.0f) && (xf < 256.0f);
    bool vy = (yf >= 0.0f) && (yf < 256.0f);
    float xc = fminf(fmaxf(xf, 0.0f), 255.0f);
    float yc = fminf(fmaxf(yf, 0.0f), 255.0f);
    int xi = (int)xc;
    int yi = (int)yc;
    float v = sm[(yi << 8) | xi];
    return (vx && vy) ? v : 0.0f;
}

// ---- bilinear sample (zero padding), matches reference _grid_sample -------
__device__ __forceinline__ float bilin(const float* sm, float ix, float iy) {
    float ix0 = floorf(ix), iy0 = floorf(iy);
    float wx = ix - ix0, wy = iy - iy0;
    float v00 = lds_tap(sm, ix0,        iy0);
    float v01 = lds_tap(sm, ix0 + 1.f,  iy0);
    float v10 = lds_tap(sm, ix0,        iy0 + 1.f);
    float v11 = lds_tap(sm, ix0 + 1.f,  iy0 + 1.f);
    float top = (1.0f - wx) * v00 + wx * v01;
    float bot = (1.0f - wx) * v10 + wx * v11;
    return (1.0f - wy) * top + wy * bot;
}

extern "C" __global__ __launch_bounds__(BLOCK)
void PartialRadonStep_kernel(const float* __restrict__ x,
                             const int* __restrict__ idxp,
                             float* __restrict__ out) {
    extern __shared__ float smem[];   // 256*256 f32 = 256 KB (of 320 KB WGP LDS)

    const int n   = blockIdx.y;
    const int tid = threadIdx.x;

    // ================= stage image n into LDS via async global->LDS ========
    {
        float* gbase = (float*)(x + (size_t)n * RW * RW);   // drop const for builtin
#if __has_builtin(__builtin_amdgcn_global_load_async_to_lds_b128)
        g4_t* g = (g4_t*)gbase;
        l4_t* l = (l4_t*)smem;
        #pragma unroll 4
        for (int i = 0; i < 64; ++i) {
            int e = i * BLOCK + tid;            // 16-byte element index, coalesced
            __builtin_amdgcn_global_load_async_to_lds_b128(g + e, l + e, 0, 0);
        }
#else
        unsigned long long gaddr = (unsigned long long)gbase;
        unsigned ldsBase = (unsigned)(unsigned long long)(lch_t*)smem;
        for (int i = 0; i < 64; ++i) {
            unsigned off = (unsigned)((i * BLOCK + tid) * 16);
            asm volatile("global_load_async_to_lds_b128 %0, %1, %2"
                         :: "v"(ldsBase + off), "v"(off), "s"(gaddr)
                         : "memory");
        }
#endif
#if __has_builtin(__builtin_amdgcn_s_wait_asynccnt)
        __builtin_amdgcn_s_wait_asynccnt(0);
#else
        asm volatile("s_wait_asynccnt 0" ::: "memory");
#endif
    }
    __syncthreads();

    // ================= per-wave work assignment ============================
    const int lane     = tid & 31;
    const int wv       = tid >> 5;        // 0..7
    const int angLocal = wv >> 1;         // 0..3
    const int colHalf  = wv & 1;          // 0..1  (128 columns each)
    const int a        = blockIdx.x * ANG_PER_BLOCK + angLocal;  // 0..179
    const int half     = lane >> 4;       // K-half for WMMA A/B layout
    const int cIdx     = lane & 15;

    // segments = linspace(1,128,6).astype(int)
    int index = idxp[0];
    index = index < 0 ? 0 : (index > 4 ? 4 : index);
    const int seg[6] = {1, 26, 51, 77, 102, 128};
    const int t0 = seg[4 - index];
    const int t1 = seg[5 - index];
    const int rInLo = 128 - t0, rInHi = 127 + t0;   // inner band (2*t0 rows)
    const int rLo   = 128 - t1, rHi   = 127 + t1;   // full  band (2*t1 rows)
    const int r0base = rLo & ~3;
    const int nch    = ((rHi - r0base) >> 2) + 1;   // 4-row chunks
    const float inv0 = 1.0f / (float)(2 * t0);
    const float inv1 = 1.0f / (float)(2 * t1);

    const float th = (float)a * 0.017453292519943295f;
    const float ct = cosf(th);
    const float st = sinf(th);

    // ================= 8 tiles of 16 columns per wave ======================
    for (int tIdx = 0; tIdx < 8; ++tIdx) {
        const int   c   = (colHalf * 8 + tIdx) * 16 + cIdx;
        const float u   = (float)c + 0.5f - 128.0f;
        const float ixu = ct * u + 127.5f;      // ix = ct*u + st*v + 127.5
        const float iyu = -st * u + 127.5f;     // iy = -st*u + ct*v + 127.5

        v8f accI = {};   // inner-band accumulator (16x16 f32 D)
        v8f accF = {};   // full-band accumulator

        for (int ch = 0; ch < nch; ++ch) {
            const int rr0 = r0base + 4 * ch + 2 * half;  // this lane's K rows
            const int rr1 = rr0 + 1;
            const float v0 = (float)rr0 - 127.5f;
            const float v1 = v0 + 1.0f;

            const float s0 = bilin(smem, ixu + st * v0, iyu + ct * v0);
            const float s1 = bilin(smem, ixu + st * v1, iyu + ct * v1);

            v2f b  = {s0, s1};                     // B: 4x16 samples (K x N)
            v2f aF = {(rr0 >= rLo   && rr0 <= rHi)   ? 1.0f : 0.0f,
                      (rr1 >= rLo   && rr1 <= rHi)   ? 1.0f : 0.0f};
            v2f aI = {(rr0 >= rInLo && rr0 <= rInHi) ? 1.0f : 0.0f,
                      (rr1 >= rInLo && rr1 <= rInHi) ? 1.0f : 0.0f};

#if __has_builtin(__builtin_amdgcn_wmma_f32_16x16x4_f32)
            accI = __builtin_amdgcn_wmma_f32_16x16x4_f32(
                false, aI, false, b, (short)0, accI, false, false);
            accF = __builtin_amdgcn_wmma_f32_16x16x4_f32(
                false, aF, false, b, (short)0, accF, false, false);
#else
            accI.s0 += aI.s0 * s0 + aI.s1 * s1;   // half-partial fallback
            accF.s0 += aF.s0 * s0 + aF.s1 * s1;
#endif
        }

        float rI = accI.s0;   // WMMA: row M=0 (lanes 0-15) holds full sum
        float rF = accF.s0;
#if !__has_builtin(__builtin_amdgcn_wmma_f32_16x16x4_f32)
        rI += __shfl_xor(rI, 16, 32);             // combine K halves
        rF += __shfl_xor(rF, 16, 32);
#endif

        if (half == 0) {
            // out[n][s][c][a], shape (4, 2, 256, 180)
            const size_t o0 = (((size_t)n * 2 + 0) * 256 + (size_t)c) * NANG + a;
            const size_t o1 = (((size_t)n * 2 + 1) * 256 + (size_t)c) * NANG + a;
            out[o0] = rI * inv0;
            out[o1] = rF * inv1;
        }
    }
}

extern "C" void kernel_launch(void* const* d_in, const int* in_sizes, int n_in,
                              void* d_out, int out_size, void* d_ws, size_t ws_size,
                              hipStream_t stream) {
    (void)in_sizes; (void)n_in; (void)out_size; (void)d_ws; (void)ws_size;
    const float* x   = (const float*)d_in[0];
    const int*   idx = (const int*)d_in[1];
    float*       out = (float*)d_out;

    const size_t shmem = (size_t)RW * RW * sizeof(float);  // 256 KB dynamic LDS
    (void)hipFuncSetAttribute((const void*)PartialRadonStep_kernel,
                              hipFuncAttributeMaxDynamicSharedMemorySize,
                              (int)shmem);

    dim3 grid(NANG / ANG_PER_BLOCK, 4);   // (45, 4)
    dim3 block(BLOCK);
    PartialRadonStep_kernel<<<grid, block, shmem, stream>>>(x, idx, out);
}